// TrafficGAT_6854767805254
// MI455X (gfx1250) — compile-verified
//
#include <hip/hip_runtime.h>
#include <math.h>

#define NEG_SLOPE 0.2f
#define EPSV 1e-16f

typedef float v2f __attribute__((ext_vector_type(2)));
typedef float v8f __attribute__((ext_vector_type(8)));

// Float atomic max via integer atomics (IEEE ordering trick); m initialized to -inf.
__device__ __forceinline__ void atomicMaxF(float* addr, float val) {
    if (val >= 0.0f) {
        atomicMax((int*)addr, __float_as_int(val));
    } else {
        atomicMin((unsigned int*)addr, __float_as_uint(val));
    }
}

// Branch-free ELU using native v_exp_f32 (argument clamped to <=0 so __expf stays in range).
__device__ __forceinline__ float elu_fast(float v) {
    float neg = __expf(fminf(v, 0.0f)) - 1.0f;
    return v > 0.0f ? v : neg;
}

__global__ void fill_kernel(float* __restrict__ p, long n, float v) {
    long i = (long)blockIdx.x * blockDim.x + threadIdx.x;
    if (i < n) p[i] = v;
}

// Layer-1 node transform: h1[n, h*32+c] = x[n,:] . w1[h*32+c,:]; also attention dots.
__global__ void transform1_kernel(const float* __restrict__ x, const float* __restrict__ w1,
                                  const float* __restrict__ aws, const float* __restrict__ awd,
                                  float* __restrict__ h1, float* __restrict__ as1,
                                  float* __restrict__ ad1, int N) {
    int t = blockIdx.x * blockDim.x + threadIdx.x;
    if (t >= N * 4) return;
    int n = t >> 2, h = t & 3;
    float x0 = x[n * 3 + 0], x1 = x[n * 3 + 1], x2 = x[n * 3 + 2];
    float ds = 0.f, dd = 0.f;
#pragma unroll
    for (int c = 0; c < 32; ++c) {
        int row = h * 32 + c;
        float v = x0 * w1[row * 3] + x1 * w1[row * 3 + 1] + x2 * w1[row * 3 + 2];
        h1[(long)n * 128 + row] = v;
        ds += v * aws[row];
        dd += v * awd[row];
    }
    as1[t] = ds;
    ad1[t] = dd;
}

// Per-edge attention logits + leaky relu + segment max (H heads).
__global__ void edge_logits_kernel(const int* __restrict__ src, const int* __restrict__ dst,
                                   const float* __restrict__ as, const float* __restrict__ ad,
                                   float* __restrict__ ex, float* __restrict__ m, int E, int H) {
    int e = blockIdx.x * blockDim.x + threadIdx.x;
    if (e >= E) return;
    int s = src[e], d = dst[e];
    for (int h = 0; h < H; ++h) {
        float v = as[s * H + h] + ad[d * H + h];
        v = v > 0.f ? v : NEG_SLOPE * v;
        ex[(long)e * H + h] = v;
        atomicMaxF(&m[d * H + h], v);
    }
}

// ex = exp(e - m[dst]); denom[dst] += ex.   (H = 1<<logH)
__global__ void edge_exp_kernel(const int* __restrict__ dst, float* __restrict__ ex,
                                const float* __restrict__ m, float* __restrict__ denom,
                                int E, int logH) {
    int t = blockIdx.x * blockDim.x + threadIdx.x;
    int H = 1 << logH;
    if (t >= (E << logH)) return;
    int e = t >> logH;
    int h = t & (H - 1);
    int d = dst[e];
    float v = __expf(ex[t] - m[d * H + h]);  // arg <= 0 -> native exp is safe & accurate
    ex[t] = v;
    atomicAdd(&denom[d * H + h], v);
}

// out[dst, c] += (ex/(denom[dst]+eps)) * hfeat[src, c].  F = 1<<logF features, H = 1<<logH heads.
__global__ void edge_scatter_kernel(const int* __restrict__ src, const int* __restrict__ dst,
                                    const float* __restrict__ ex, const float* __restrict__ denom,
                                    const float* __restrict__ hfeat, float* __restrict__ out,
                                    long total, int logF, int logH) {
    long t = (long)blockIdx.x * blockDim.x + threadIdx.x;
    if (t >= total) return;
    int F = 1 << logF;
    long e = t >> logF;
    int c = (int)(t & (F - 1));
    int h = c >> (logF - logH);
    int s = src[e], d = dst[e];
    int H = 1 << logH;
    float alpha = ex[(e << logH) + h] / (denom[d * H + h] + EPSV);
    atomicAdd(&out[(long)d * F + c], alpha * hfeat[(long)s * F + c]);
}

// h2 = elu(out1 + b1) @ w2^T   via V_WMMA_F32_16X16X4_F32.  One wave per 16-node tile, N=32 cols.
__global__ void h2_gemm_kernel(const float* __restrict__ out1, const float* __restrict__ b1,
                               const float* __restrict__ w2, float* __restrict__ h2, int ntiles) {
    int lane = threadIdx.x & 31;
    int wave = (blockIdx.x * blockDim.x + threadIdx.x) >> 5;
    if (wave >= ntiles) return;  // wave-uniform: EXEC all-1 at WMMA
    int mbase = wave * 16;
    int m = lane & 15;
    int khalf = (lane >> 4) * 2;
    const float* arow = out1 + (long)(mbase + m) * 128;
    const float* brow0 = w2 + (long)m * 128;         // n = m
    const float* brow1 = w2 + (long)(m + 16) * 128;  // n = m+16
    v8f c0 = {};
    v8f c1 = {};
    for (int ks = 0; ks < 32; ++ks) {
        int k = ks * 4 + khalf;
        float a0 = elu_fast(arow[k] + b1[k]);
        float a1 = elu_fast(arow[k + 1] + b1[k + 1]);
        v2f A = {a0, a1};
        v2f B0 = {brow0[k], brow0[k + 1]};
        v2f B1 = {brow1[k], brow1[k + 1]};
        c0 = __builtin_amdgcn_wmma_f32_16x16x4_f32(false, A, false, B0, (short)0, c0, false, false);
        c1 = __builtin_amdgcn_wmma_f32_16x16x4_f32(false, A, false, B1, (short)0, c1, false, false);
    }
    int rofs = (lane >> 4) * 8;
#pragma unroll
    for (int r = 0; r < 8; ++r) {
        int node = mbase + r + rofs;
        h2[(long)node * 32 + m] = c0[r];
        h2[(long)node * 32 + m + 16] = c1[r];
    }
}

// Layer-2 attention dots (H=1, C=32).
__global__ void alpha2_kernel(const float* __restrict__ h2, const float* __restrict__ aws,
                              const float* __restrict__ awd, float* __restrict__ as2,
                              float* __restrict__ ad2, int N) {
    int n = blockIdx.x * blockDim.x + threadIdx.x;
    if (n >= N) return;
    float ds = 0.f, dd = 0.f;
#pragma unroll
    for (int c = 0; c < 32; ++c) {
        float v = h2[(long)n * 32 + c];
        ds += v * aws[c];
        dd += v * awd[c];
    }
    as2[n] = ds;
    ad2[n] = dd;
}

// Edge MLP: ef = [h[src]+b2, h[dst]+b2] (64) -> relu(ef @ w1m^T + b1m) (32) -> relu(. w2m^T + b2m).
// One wave per 16-edge tile; layer-1 via WMMA f32 16x16x4, layer-2 via shfl reduction.
__global__ void mlp_kernel(const int* __restrict__ src, const int* __restrict__ dst,
                           const float* __restrict__ out2, const float* __restrict__ b2,
                           const float* __restrict__ w1m, const float* __restrict__ b1m,
                           const float* __restrict__ w2m, const float* __restrict__ b2m,
                           float* __restrict__ flows, int etiles) {
    int lane = threadIdx.x & 31;
    int wave = (blockIdx.x * blockDim.x + threadIdx.x) >> 5;
    if (wave >= etiles) return;  // wave-uniform
    int ebase = wave * 16;
    int m = lane & 15;
    int e = ebase + m;
    int s = src[e], d = dst[e];
    const float* rs = out2 + (long)s * 32;
    const float* rd = out2 + (long)d * 32;
    int khalf = (lane >> 4) * 2;
    const float* brow0 = w1m + (long)m * 64;
    const float* brow1 = w1m + (long)(m + 16) * 64;
    v8f c0 = {};
    v8f c1 = {};
    for (int ks = 0; ks < 16; ++ks) {
        int k = ks * 4 + khalf;
        // k and k+1 always lie on the same side of 32 (k even) -> one base select, one b64 load
        const float* rbase = (k < 32) ? rs : rd;
        int kk = k & 31;
        float a0 = rbase[kk] + b2[kk];
        float a1 = rbase[kk + 1] + b2[kk + 1];
        v2f A = {a0, a1};
        v2f B0 = {brow0[k], brow0[k + 1]};
        v2f B1 = {brow1[k], brow1[k + 1]};
        c0 = __builtin_amdgcn_wmma_f32_16x16x4_f32(false, A, false, B0, (short)0, c0, false, false);
        c1 = __builtin_amdgcn_wmma_f32_16x16x4_f32(false, A, false, B1, (short)0, c1, false, false);
    }
    float bias0 = b1m[m], bias1 = b1m[m + 16];
    float wa = w2m[m], wb = w2m[m + 16];
    float outb = b2m[0];
    int rofs = (lane >> 4) * 8;
#pragma unroll
    for (int r = 0; r < 8; ++r) {
        float v0 = c0[r] + bias0;
        v0 = v0 > 0.f ? v0 : 0.f;
        float v1 = c1[r] + bias1;
        v1 = v1 > 0.f ? v1 : 0.f;
        float p = v0 * wa + v1 * wb;
        p += __shfl_xor(p, 1);
        p += __shfl_xor(p, 2);
        p += __shfl_xor(p, 4);
        p += __shfl_xor(p, 8);
        if (m == 0) {
            float f = p + outb;
            flows[ebase + r + rofs] = f > 0.f ? f : 0.f;
        }
    }
}

extern "C" void kernel_launch(void* const* d_in, const int* in_sizes, int n_in,
                              void* d_out, int out_size, void* d_ws, size_t ws_size,
                              hipStream_t stream) {
    const float* x    = (const float*)d_in[0];
    const int*   ei   = (const int*)d_in[1];
    const float* w1   = (const float*)d_in[2];
    const float* aws1 = (const float*)d_in[3];
    const float* awd1 = (const float*)d_in[4];
    const float* b1   = (const float*)d_in[5];
    const float* w2   = (const float*)d_in[6];
    const float* aws2 = (const float*)d_in[7];
    const float* awd2 = (const float*)d_in[8];
    const float* b2   = (const float*)d_in[9];
    const float* mw1  = (const float*)d_in[10];
    const float* mb1  = (const float*)d_in[11];
    const float* mw2  = (const float*)d_in[12];
    const float* mb2  = (const float*)d_in[13];
    float* flows = (float*)d_out;

    int N = in_sizes[0] / 3;   // 100000
    int E = in_sizes[1] / 2;   // 1600000
    const int* src = ei;
    const int* dst = ei + E;

    // Workspace carving (floats). Zero-init region laid out contiguously, -inf region likewise.
    float* p = (float*)d_ws;
    float* out1   = p; p += (size_t)N * 128;
    float* out2   = p; p += (size_t)N * 32;
    float* denom1 = p; p += (size_t)N * 4;
    float* denom2 = p; p += (size_t)N;
    size_t zeroCount = (size_t)N * 165;
    float* m1 = p; p += (size_t)N * 4;
    float* m2 = p; p += (size_t)N;
    size_t ninfCount = (size_t)N * 5;
    float* h1  = p; p += (size_t)N * 128;
    float* h2  = p; p += (size_t)N * 32;
    float* as1 = p; p += (size_t)N * 4;
    float* ad1 = p; p += (size_t)N * 4;
    float* as2 = p; p += (size_t)N;
    float* ad2 = p; p += (size_t)N;
    float* ex1 = p; p += (size_t)E * 4;
    float* ex2 = p; p += (size_t)E;

    const int B = 256;
    fill_kernel<<<(int)((zeroCount + B - 1) / B), B, 0, stream>>>(out1, (long)zeroCount, 0.0f);
    fill_kernel<<<(int)((ninfCount + B - 1) / B), B, 0, stream>>>(m1, (long)ninfCount, -INFINITY);

    // ---- GAT layer 1 (H=4, C=32) ----
    transform1_kernel<<<(N * 4 + B - 1) / B, B, 0, stream>>>(x, w1, aws1, awd1, h1, as1, ad1, N);
    edge_logits_kernel<<<(E + B - 1) / B, B, 0, stream>>>(src, dst, as1, ad1, ex1, m1, E, 4);
    edge_exp_kernel<<<(E * 4 + B - 1) / B, B, 0, stream>>>(dst, ex1, m1, denom1, E, 2);
    long tot1 = (long)E * 128;
    edge_scatter_kernel<<<(int)((tot1 + B - 1) / B), B, 0, stream>>>(src, dst, ex1, denom1, h1,
                                                                     out1, tot1, 7, 2);

    // ---- h2 = elu(out1+b1) @ w2^T  (WMMA) ----
    int ntiles = N / 16;  // 6250
    h2_gemm_kernel<<<(ntiles + 9) / 10, 320, 0, stream>>>(out1, b1, w2, h2, ntiles);

    // ---- GAT layer 2 (H=1, C=32) ----
    alpha2_kernel<<<(N + B - 1) / B, B, 0, stream>>>(h2, aws2, awd2, as2, ad2, N);
    edge_logits_kernel<<<(E + B - 1) / B, B, 0, stream>>>(src, dst, as2, ad2, ex2, m2, E, 1);
    edge_exp_kernel<<<(E + B - 1) / B, B, 0, stream>>>(dst, ex2, m2, denom2, E, 0);
    long tot2 = (long)E * 32;
    edge_scatter_kernel<<<(int)((tot2 + B - 1) / B), B, 0, stream>>>(src, dst, ex2, denom2, h2,
                                                                     out2, tot2, 5, 0);

    // ---- Edge MLP (WMMA) -> flows ----
    int etiles = E / 16;  // 100000
    mlp_kernel<<<(etiles + 9) / 10, 320, 0, stream>>>(src, dst, out2, b2, mw1, mb1, mw2, mb2,
                                                      flows, etiles);
}